// MultiheadAttentionRelation_42700564857532
// MI455X (gfx1250) — compile-verified
//
#include <hip/hip_runtime.h>
#include <hip/hip_bf16.h>
#include <math.h>

#define B_  2
#define L_  256
#define E_  512
#define H_  8
#define HD_ 64

typedef __attribute__((ext_vector_type(2))) float v2f;
typedef __attribute__((ext_vector_type(8))) float v8f;

// D = A(16x4) * B(4x16) + C, fp32 WMMA (V_WMMA_F32_16X16X4_F32)
__device__ __forceinline__ v8f wmma_f32x4(v2f a, v2f b, v8f c) {
  return __builtin_amdgcn_wmma_f32_16x16x4_f32(
      /*neg_a=*/false, a, /*neg_b=*/false, b,
      /*c_mod=*/(short)0, c, /*reuse_a=*/false, /*reuse_b=*/false);
}

// ---------------------------------------------------------------------------
// Generic LDS-tiled fp32 GEMM: C[M,N] = A[M,K](lda) @ op(B)(ldb) (+ bias)
// 64x64 tile / workgroup of 256 threads, 4x4 micro-tile per thread.
// All dims used here divide evenly (M,N mult of 64; K mult of 16).
// ---------------------------------------------------------------------------
__global__ __launch_bounds__(256) void gemm64(
    const float* __restrict__ A, int lda,
    const float* __restrict__ Bm, int ldb, int transB,
    float* __restrict__ C, int ldc,
    const float* __restrict__ bias, int K)
{
  __shared__ float As[64][17];
  __shared__ float Bs[16][68];
  const int tx = threadIdx.x & 15;
  const int ty = threadIdx.x >> 4;
  const int row0 = blockIdx.x * 64;
  const int col0 = blockIdx.y * 64;

  float acc[4][4] = {};

  for (int k0 = 0; k0 < K; k0 += 16) {
    for (int i = threadIdx.x; i < 64 * 16; i += 256) {
      int r = i >> 4, kk = i & 15;
      As[r][kk] = A[(size_t)(row0 + r) * lda + k0 + kk];
    }
    for (int i = threadIdx.x; i < 16 * 64; i += 256) {
      int kk = i >> 6, c = i & 63;
      Bs[kk][c] = transB ? Bm[(size_t)(col0 + c) * ldb + k0 + kk]
                         : Bm[(size_t)(k0 + kk) * ldb + col0 + c];
    }
    __syncthreads();
#pragma unroll
    for (int kk = 0; kk < 16; ++kk) {
      float bv[4];
#pragma unroll
      for (int j = 0; j < 4; ++j) bv[j] = Bs[kk][tx * 4 + j];
#pragma unroll
      for (int i2 = 0; i2 < 4; ++i2) {
        float av = As[ty * 4 + i2][kk];
#pragma unroll
        for (int j = 0; j < 4; ++j) acc[i2][j] += av * bv[j];
      }
    }
    __syncthreads();
  }

  for (int i2 = 0; i2 < 4; ++i2)
    for (int j = 0; j < 4; ++j) {
      int r = row0 + ty * 4 + i2, c = col0 + tx * 4 + j;
      float v = acc[i2][j];
      if (bias) v += bias[c];
      C[(size_t)r * ldc + c] = v;
    }
}

// ---------------------------------------------------------------------------
// brdot[b,k,h] = br[h*64:..] . khat[b,k,h*64:..]
// ---------------------------------------------------------------------------
__global__ void brdot_kernel(const float* __restrict__ khat,
                             const float* __restrict__ br,
                             float* __restrict__ brdot)
{
  int idx = blockIdx.x * blockDim.x + threadIdx.x;  // (b*L + k)*H + h
  if (idx >= B_ * L_ * H_) return;
  int h = idx & (H_ - 1);
  int bk = idx >> 3;
  const float* kp = khat + (size_t)bk * E_ + h * HD_;
  const float* bp = br + h * HD_;
  float s = 0.f;
  for (int j = 0; j < HD_; ++j) s += kp[j] * bp[j];
  brdot[idx] = s;
}

// ---------------------------------------------------------------------------
// qk logits (unscaled) + br.khat term: lg[b][q][h][k]
// one wave per 16x16 (q,k) tile of one (b,h); WMMA f32 16x16x4 over K=64.
// ---------------------------------------------------------------------------
__global__ __launch_bounds__(32) void qk_wmma(
    const float* __restrict__ qhat, const float* __restrict__ khat,
    const float* __restrict__ brdot, float* __restrict__ lg)
{
  const int q0 = blockIdx.x * 16;
  const int k0 = blockIdx.y * 16;
  const int b  = blockIdx.z >> 3;
  const int h  = blockIdx.z & 7;
  const int lane = threadIdx.x;
  const int lo = lane & 15, hi = lane >> 4;

  // A[M=q, K=e] : lanes0-15 K={e,e+1}; lanes16-31 K={e+2,e+3}
  const float* Ap = qhat + (size_t)(b * L_ + q0 + lo) * E_ + h * HD_ + hi * 2;
  // B[K=e, N=k] : same per-lane pair pattern, lane selects column k0+lo
  const float* Bp = khat + (size_t)(b * L_ + k0 + lo) * E_ + h * HD_ + hi * 2;

  v8f c = {};
#pragma unroll
  for (int e = 0; e < HD_; e += 4) {
    v2f a = *(const v2f*)(Ap + e);
    v2f bb = *(const v2f*)(Bp + e);
    c = wmma_f32x4(a, bb, c);
  }

  float bd = brdot[(size_t)(b * L_ + k0 + lo) * H_ + h];
#pragma unroll
  for (int r = 0; r < 8; ++r) {
    int q = q0 + r + 8 * hi;
    lg[((size_t)(b * L_ + q) * H_ + h) * L_ + k0 + lo] = c[r] + bd;
  }
}

// ---------------------------------------------------------------------------
// Hot kernel: stream relation exactly once (256 MB -> HBM roofline).
// One workgroup (16 waves) per (b,k):
//   - stage U[b,k] (16x512, h-padded) in LDS once (pitch 516: conflict-free)
//   - each wave owns one 16-row q-tile; per 64-float e-chunk it stages its
//     relation rows via full-cacheline global_load_b128 into a private LDS
//     region (pitch 68: WMMA ds_load_b64 pattern hits all 64 banks), then
//     runs 16 WMMA k-steps from LDS.
// LDS ops within a wave are architecturally in-order (DScnt), so the
// store->load RAW and the chunk-reuse WAR need no explicit waits.
//   lg[b][q][h][k] += relation[b,q,k,:] . U[b,k,h,:]
// ---------------------------------------------------------------------------
#define UPITCH 516  // 512 + 4 pad
#define APITCH 68   // 64 + 4 pad

__global__ __launch_bounds__(512) void rel_wmma(
    const float* __restrict__ relation, const float* __restrict__ Upad,
    float* __restrict__ lg)
{
  __shared__ float Us[16 * UPITCH];          // 33 KB
  __shared__ float As[16 * 16 * APITCH];     // 68 KB (16 waves x 16 rows x 68)
  const int kk = blockIdx.x;
  const int b  = blockIdx.y;
  const int t  = threadIdx.x;

  // cooperative load of the 16x512 U slice
  const float* up = Upad + (size_t)(b * L_ + kk) * 16 * E_;
  for (int i = t * 4; i < 16 * E_; i += 512 * 4) {
    float4 v = *(const float4*)(up + i);
    int hp = i >> 9, e = i & 511;
    *(float4*)(&Us[hp * UPITCH + e]) = v;
  }
  __syncthreads();

  const int wave = t >> 5;
  const int lane = t & 31;
  const int lo = lane & 15, hi = lane >> 4;
  const int q0 = wave * 16;
  const int h = lo;  // output column = head (only 0..7 valid)

  // C init = qk logits tile
  v8f c = {};
  if (h < H_) {
#pragma unroll
    for (int r = 0; r < 8; ++r) {
      int q = q0 + r + 8 * hi;
      c[r] = lg[((size_t)(b * L_ + q) * H_ + h) * L_ + kk];
    }
  }

  // staging assignment: 4 lanes-groups x 8 lanes -> 4 rows x 128B per request
  const int r4 = lane >> 3;  // 0..3
  const int c8 = lane & 7;   // 0..7
  float* aw = &As[wave * 16 * APITCH];
  const size_t rowstride = (size_t)L_ * E_;  // between q rows of relation
  const float* Ag = relation +
      ((size_t)(b * L_ + q0 + r4) * L_ + kk) * E_ + c8 * 4;

  const float* Als = &aw[lo * APITCH + hi * 2];
  const float* Bls = &Us[lo * UPITCH + hi * 2];

  for (int ec = 0; ec < E_; ec += 64) {
    // stage 16 rows x 64 floats: 8 full-cacheline b128 loads per lane-set
#pragma unroll
    for (int g = 0; g < 4; ++g) {
#pragma unroll
      for (int half = 0; half < 2; ++half) {
        float4 v = *(const float4*)(Ag + (size_t)(4 * g) * rowstride +
                                    ec + 32 * half);
        *(float4*)(&aw[(r4 + 4 * g) * APITCH + c8 * 4 + 32 * half]) = v;
      }
    }
    // consume: 16 WMMA k-steps fed from LDS
#pragma unroll
    for (int e2 = 0; e2 < 64; e2 += 4) {
      v2f a = *(const v2f*)(Als + e2);
      v2f bb = *(const v2f*)(Bls + ec + e2);
      c = wmma_f32x4(a, bb, c);
    }
  }

  if (h < H_) {
#pragma unroll
    for (int r = 0; r < 8; ++r) {
      int q = q0 + r + 8 * hi;
      lg[((size_t)(b * L_ + q) * H_ + h) * L_ + kk] = c[r];
    }
  }
}

// ---------------------------------------------------------------------------
// Softmax over k (scale 1/8 applied here). One wave per (b,q,h) row.
// Writes a_out in (B,L,L,H) layout and a2 in (B,H,L,L) (k-contiguous).
// ---------------------------------------------------------------------------
__global__ __launch_bounds__(256) void softmax_kernel(
    const float* __restrict__ lg, float* __restrict__ a_out,
    float* __restrict__ a2)
{
  const int row = blockIdx.x * 8 + (threadIdx.x >> 5);  // (b*L+q)*H + h
  const int lane = threadIdx.x & 31;
  const int h = row & (H_ - 1);
  const int bq = row >> 3;
  const int b = bq >> 8;       // L_ == 256
  const int q = bq & (L_ - 1);

  const float* rp = lg + (size_t)row * L_;
  float x[8];
  float m = -INFINITY;
#pragma unroll
  for (int i = 0; i < 8; ++i) {
    x[i] = rp[lane + 32 * i];
    m = fmaxf(m, x[i]);
  }
  for (int off = 16; off >= 1; off >>= 1) m = fmaxf(m, __shfl_xor(m, off, 32));

  const float scale = 0.125f;  // 1/sqrt(64)
  float s = 0.f;
#pragma unroll
  for (int i = 0; i < 8; ++i) {
    x[i] = __expf((x[i] - m) * scale);
    s += x[i];
  }
  for (int off = 16; off >= 1; off >>= 1) s += __shfl_xor(s, off, 32);
  const float inv = 1.0f / s;

#pragma unroll
  for (int i = 0; i < 8; ++i) {
    int k = lane + 32 * i;
    float v = x[i] * inv;
    a_out[((size_t)(b * L_ + q) * L_ + k) * H_ + h] = v;
    a2[((size_t)(b * H_ + h) * L_ + q) * L_ + k] = v;
  }
}

// ---------------------------------------------------------------------------
// attn[b,q,h,:] = sum_k a2[b,h,q,k] * vhat[b,k,h,:]  (WMMA, per (b,h))
// writes directly into (B,L,E) layout.
// ---------------------------------------------------------------------------
__global__ __launch_bounds__(32) void attn_wmma(
    const float* __restrict__ a2, const float* __restrict__ vhat,
    float* __restrict__ attnbuf)
{
  const int q0 = blockIdx.x * 16;
  const int e0 = blockIdx.y * 16;
  const int b  = blockIdx.z >> 3;
  const int h  = blockIdx.z & 7;
  const int lane = threadIdx.x;
  const int lo = lane & 15, hi = lane >> 4;

  // A[M=q, K=k] = a2[b,h,q,k] (k-contiguous float2 per lane)
  const float* Ap = a2 + ((size_t)(b * H_ + h) * L_ + q0 + lo) * L_ + hi * 2;
  // B[K=k, N=e] = vhat[b,k, h*64 + e]; rows stride E_
  const float* Bp = vhat + (size_t)(b * L_ + hi * 2) * E_ + h * HD_ + e0 + lo;

  v8f c = {};
  for (int k = 0; k < L_; k += 4) {
    v2f a = *(const v2f*)(Ap + k);
    v2f bb;
    bb.x = Bp[(size_t)k * E_];        // row k + hi*2
    bb.y = Bp[(size_t)(k + 1) * E_];  // row k + hi*2 + 1
    c = wmma_f32x4(a, bb, c);
  }

#pragma unroll
  for (int r = 0; r < 8; ++r) {
    int q = q0 + r + 8 * hi;
    attnbuf[(size_t)(b * L_ + q) * E_ + h * HD_ + e0 + lo] = c[r];
  }
}

// ---------------------------------------------------------------------------
extern "C" void kernel_launch(void* const* d_in, const int* in_sizes, int n_in,
                              void* d_out, int out_size, void* d_ws, size_t ws_size,
                              hipStream_t stream) {
  const float* query    = (const float*)d_in[0];
  const float* key      = (const float*)d_in[1];
  const float* value    = (const float*)d_in[2];
  const float* relation = (const float*)d_in[3];
  const float* Wq       = (const float*)d_in[4];
  const float* Wk       = (const float*)d_in[5];
  // d_in[6] (Wv) intentionally unused: reference projects value with Wq (bug preserved)
  const float* Wr       = (const float*)d_in[7];
  const float* br       = (const float*)d_in[8];
  const float* Wo       = (const float*)d_in[9];
  const float* bo       = (const float*)d_in[10];

  float* out_ptr = (float*)d_out;                        // (B,L,E)
  float* a_ptr   = (float*)d_out + (size_t)B_ * L_ * E_; // (B,L,L,H)

  float* ws = (float*)d_ws;
  float* qhat  = ws;                      // 262144
  float* khat  = ws + 262144;             // 262144
  float* vhat  = ws + 524288;             // 262144
  float* attnb = ws + 786432;             // 262144
  float* brd   = ws + 1048576;            // 4096
  float* lg    = ws + 1052672;            // 1048576  lg[b][q][h][k]
  float* a2    = ws + 2101248;            // 1048576  a2[b][h][q][k]
  float* Upad  = ws + 3149824;            // 4194304  U[b][k][16][E]
  (void)in_sizes; (void)n_in; (void)out_size; (void)ws_size;

  const int BL = B_ * L_;  // 512

  // zero the h=8..15 padding rows of U (garbage-free B operand)
  hipMemsetAsync(Upad, 0, (size_t)4194304 * sizeof(float), stream);

  dim3 g88(BL / 64, E_ / 64);
  // projections: qhat = query@Wq, khat = key@Wk, vhat = value@Wq (bug)
  gemm64<<<g88, 256, 0, stream>>>(query, E_, Wq, E_, 0, qhat, E_, nullptr, E_);
  gemm64<<<g88, 256, 0, stream>>>(key,   E_, Wk, E_, 0, khat, E_, nullptr, E_);
  gemm64<<<g88, 256, 0, stream>>>(value, E_, Wq, E_, 0, vhat, E_, nullptr, E_);

  // brdot[b,k,h] = br . khat slice
  brdot_kernel<<<(B_ * L_ * H_ + 255) / 256, 256, 0, stream>>>(khat, br, brd);

  // U_h = khat[:, h*64:(h+1)*64] @ Wr[h*64:(h+1)*64, :]  ->  Upad[:, h, :]
  for (int h = 0; h < H_; ++h) {
    gemm64<<<g88, 256, 0, stream>>>(khat + h * HD_, E_,
                                    Wr + (size_t)h * HD_ * E_, E_, 0,
                                    Upad + (size_t)h * E_, 16 * E_, nullptr, HD_);
  }

  // qk logits (+ br term), unscaled
  qk_wmma<<<dim3(L_ / 16, L_ / 16, B_ * H_), 32, 0, stream>>>(qhat, khat, brd, lg);

  // hot pass: accumulate relation . U onto logits (streams 256MB once)
  rel_wmma<<<dim3(L_, B_), 512, 0, stream>>>(relation, Upad, lg);

  // softmax over k, scale 1/8; writes `a` output + k-contiguous copy
  softmax_kernel<<<(B_ * L_ * H_) / 8, 256, 0, stream>>>(lg, a_ptr, a2);

  // attn = a @ vhat per (b,h)
  attn_wmma<<<dim3(L_ / 16, HD_ / 16, B_ * H_), 32, 0, stream>>>(a2, vhat, attnb);

  // out = attn @ Wo^T + bo
  gemm64<<<g88, 256, 0, stream>>>(attnb, E_, Wo, E_, 1, out_ptr, E_, bo, E_);
}